// Net_44014824849908
// MI455X (gfx1250) — compile-verified
//
#include <hip/hip_runtime.h>
#include <hip/hip_bf16.h>
#include <cstddef>

// Problem constants (match reference)
#define NN 50000
#define CC 2000
#define DD 128
#define KK 8
#define UU 20000
#define QG 80      // partial-Q accumulator groups

typedef float v2f __attribute__((ext_vector_type(2)));
typedef float v8f __attribute__((ext_vector_type(8)));

// ---------------------------------------------------------------------------
// K1: W[c,d,k] = softmax over d of CA[c,d,k].   grid=C, block=128 (d)
// ---------------------------------------------------------------------------
__global__ void k_softmax_CA(const float* __restrict__ CA, float* __restrict__ W) {
    __shared__ float buf[DD * KK];
    __shared__ float mk[KK], ik[KK];
    const int c = blockIdx.x, d = threadIdx.x;
    const float* src = CA + (size_t)c * (DD * KK) + (size_t)d * KK;
#pragma unroll
    for (int k = 0; k < KK; ++k) buf[d * KK + k] = src[k];
    __syncthreads();
    if (d < KK) {
        const int k = d;
        float m = -3.402823466e38f;
        for (int j = 0; j < DD; ++j) m = fmaxf(m, buf[j * KK + k]);
        float s = 0.0f;
        for (int j = 0; j < DD; ++j) s += __expf(buf[j * KK + k] - m);
        mk[k] = m; ik[k] = 1.0f / s;
    }
    __syncthreads();
    float* dst = W + (size_t)c * (DD * KK) + (size_t)d * KK;
#pragma unroll
    for (int k = 0; k < KK; ++k) dst[k] = __expf(buf[d * KK + k] - mk[k]) * ik[k];
}

// ---------------------------------------------------------------------------
// K2: A[n,k] = sum_d X[n,d] * W[cour[n],d,k].   grid=N, block=128 (d)
// ---------------------------------------------------------------------------
__global__ void k_compute_A(const float* __restrict__ X, const int* __restrict__ cour,
                            const float* __restrict__ W, float* __restrict__ A) {
    __shared__ float buf[DD * KK];
    const int n = blockIdx.x, d = threadIdx.x;
    const int c = cour[n];
    const float x = X[(size_t)n * DD + d];
    const float* w = W + (size_t)c * (DD * KK) + (size_t)d * KK;
#pragma unroll
    for (int k = 0; k < KK; ++k) buf[d * KK + k] = x * w[k];
    __syncthreads();
    for (int stride = DD / 2; stride >= 1; stride >>= 1) {
        if (d < stride) {
#pragma unroll
            for (int k = 0; k < KK; ++k) buf[d * KK + k] += buf[(d + stride) * KK + k];
        }
        __syncthreads();
    }
    if (d < KK) A[(size_t)n * KK + d] = buf[d];
}

// ---------------------------------------------------------------------------
// K3: UAs = softmax(UA, axis=0).   grid=K, block=256 (strided over U)
// ---------------------------------------------------------------------------
__global__ void k_softmax_UA(const float* __restrict__ UA, float* __restrict__ UAs) {
    __shared__ float red[256];
    const int k = blockIdx.x, t = threadIdx.x;
    float m = -3.402823466e38f;
    for (int u = t; u < UU; u += 256) m = fmaxf(m, UA[(size_t)u * KK + k]);
    red[t] = m; __syncthreads();
    for (int s = 128; s >= 1; s >>= 1) { if (t < s) red[t] = fmaxf(red[t], red[t + s]); __syncthreads(); }
    m = red[0]; __syncthreads();
    float sum = 0.0f;
    for (int u = t; u < UU; u += 256) sum += __expf(UA[(size_t)u * KK + k] - m);
    red[t] = sum; __syncthreads();
    for (int s = 128; s >= 1; s >>= 1) { if (t < s) red[t] += red[t + s]; __syncthreads(); }
    const float inv = 1.0f / red[0];
    for (int u = t; u < UU; u += 256) UAs[(size_t)u * KK + k] = __expf(UA[(size_t)u * KK + k] - m) * inv;
}

// ---------------------------------------------------------------------------
// K4: per-row max and 1/sum(exp) of UC.   grid=U, block=256
// ---------------------------------------------------------------------------
__global__ void k_row_stats_UC(const float* __restrict__ UC,
                               float* __restrict__ rowmax, float* __restrict__ rowinv) {
    __shared__ float red[256];
    const int u = blockIdx.x, t = threadIdx.x;
    const float* row = UC + (size_t)u * CC;
    float m = -3.402823466e38f;
    for (int c = t; c < CC; c += 256) m = fmaxf(m, row[c]);
    red[t] = m; __syncthreads();
    for (int s = 128; s >= 1; s >>= 1) { if (t < s) red[t] = fmaxf(red[t], red[t + s]); __syncthreads(); }
    m = red[0]; __syncthreads();
    float sum = 0.0f;
    for (int c = t; c < CC; c += 256) sum += __expf(row[c] - m);
    red[t] = sum; __syncthreads();
    for (int s = 128; s >= 1; s >>= 1) { if (t < s) red[t] += red[t + s]; __syncthreads(); }
    if (t == 0) { rowmax[u] = m; rowinv[u] = 1.0f / red[0]; }
}

// ---------------------------------------------------------------------------
// K5: deterministic partial Q. Block g handles a u-chunk; thread t owns
//     c = t + 256*j (j<8), keeping 8x8 register accumulators.
// ---------------------------------------------------------------------------
__global__ void __launch_bounds__(256) k_compute_Qpart(
        const float* __restrict__ UC, const float* __restrict__ UAs,
        const float* __restrict__ rowmax, const float* __restrict__ rowinv,
        float* __restrict__ Qpart) {
    const int g = blockIdx.x, t = threadIdx.x;
    const int chunk = (UU + QG - 1) / QG;
    const int u0 = g * chunk;
    const int u1 = (u0 + chunk < UU) ? (u0 + chunk) : UU;
    float acc[8][KK];
#pragma unroll
    for (int j = 0; j < 8; ++j)
#pragma unroll
        for (int k = 0; k < KK; ++k) acc[j][k] = 0.0f;

    for (int u = u0; u < u1; ++u) {
        const float m = rowmax[u];
        const float iv = rowinv[u];
        float ua[KK];
#pragma unroll
        for (int k = 0; k < KK; ++k) ua[k] = UAs[(size_t)u * KK + k] * iv;
        const float* row = UC + (size_t)u * CC;
#pragma unroll
        for (int j = 0; j < 8; ++j) {
            const int c = t + 256 * j;
            if (c < CC) {
                const float e = __expf(row[c] - m);
#pragma unroll
                for (int k = 0; k < KK; ++k) acc[j][k] += e * ua[k];
            }
        }
    }
    float* qp = Qpart + (size_t)g * (CC * KK);
#pragma unroll
    for (int j = 0; j < 8; ++j) {
        const int c = t + 256 * j;
        if (c < CC) {
#pragma unroll
            for (int k = 0; k < KK; ++k) qp[(size_t)c * KK + k] = acc[j][k];
        }
    }
}

// ---------------------------------------------------------------------------
// K6: Q = sum_g Qpart[g]
// ---------------------------------------------------------------------------
__global__ void k_reduce_Q(const float* __restrict__ Qpart, float* __restrict__ Q) {
    const int idx = blockIdx.x * 256 + threadIdx.x;
    if (idx >= CC * KK) return;
    float s = 0.0f;
    for (int g = 0; g < QG; ++g) s += Qpart[(size_t)g * (CC * KK) + idx];
    Q[idx] = s;
}

// ---------------------------------------------------------------------------
// K7: alpha[c] = 1 - sigmoid(slide) - sigmoid(guess); beta[c] = sigmoid(guess)
// ---------------------------------------------------------------------------
__global__ void k_epilogue_consts(const float* __restrict__ guess_, const float* __restrict__ slide_,
                                  float* __restrict__ alpha, float* __restrict__ beta) {
    const int c = blockIdx.x * 256 + threadIdx.x;
    if (c >= CC) return;
    const float gg = 1.0f / (1.0f + __expf(-guess_[c]));
    const float sl = 1.0f / (1.0f + __expf(-slide_[c]));
    alpha[c] = 1.0f - sl - gg;
    beta[c] = gg;
}

// ---------------------------------------------------------------------------
// K8: Y = alpha[c] * (A @ Q^T) + beta[c]  via V_WMMA_F32_16X16X4_F32.
// Main kernel: 781 FULL 64-row M-tiles x 125 16-col C-tiles. One wave per
// (mtile,ctile): B fragments loaded once, 4 sub-tiles x 2 chained WMMAs,
// completely unguarded NT stores (constant byte offsets).
// Tail kernel: rows 49984..49999 (exactly one 16-row tile) x 125 ctiles,
// also guard-free. Tile id scalarized via readfirstlane -> SALU control flow.
// ---------------------------------------------------------------------------
#define CTILES (CC / 16)                       // 125
#define MTILES_FULL (NN / 64)                  // 781 (full 64-row tiles)
#define TOTAL_FULL (CTILES * MTILES_FULL)      // 97625
#define TAIL_ROW0 (MTILES_FULL * 64)           // 49984; NN - TAIL_ROW0 == 16

__device__ __forceinline__ void wmma_tile16(
        const float* __restrict__ A, const float* __restrict__ Q_unused,
        int r0, int lo, int hi, v2f b0, v2f b1,
        float al, float be, float* __restrict__ Y, int col) {
    const float* ap = A + (size_t)(r0 + lo) * KK + hi * 2;
    v2f a0, a1;
    a0.x = ap[0]; a0.y = ap[1];
    a1.x = ap[4]; a1.y = ap[5];
    v8f d = {};
    d = __builtin_amdgcn_wmma_f32_16x16x4_f32(false, a0, false, b0, (short)0, d, false, false);
    d = __builtin_amdgcn_wmma_f32_16x16x4_f32(false, a1, false, b1, (short)0, d, false, false);
    float* yp = Y + (size_t)(r0 + (hi << 3)) * CC + col;
#pragma unroll
    for (int j = 0; j < 8; ++j)
        __builtin_nontemporal_store(al * d[j] + be, yp + (size_t)j * CC);
}

__global__ void __launch_bounds__(256) k_gemm_Y(
        const float* __restrict__ A, const float* __restrict__ Q,
        const float* __restrict__ alpha, const float* __restrict__ beta,
        float* __restrict__ Y) {
    // scalarize tile id: wave-uniform by construction -> force into SGPR
    const int wave = __builtin_amdgcn_readfirstlane((blockIdx.x << 3) + (threadIdx.x >> 5));
    if (wave >= TOTAL_FULL) return;            // scalar branch
    const int lane = threadIdx.x & 31;
    const int mtile = wave / CTILES;
    const int ctile = wave - mtile * CTILES;
    const int c0 = ctile << 4;
    const int row0 = mtile << 6;
    const int lo = lane & 15;
    const int hi = lane >> 4;                  // 0 or 1
    const int col = c0 + lo;

    // B = Q^T (4x16 per fragment): lane<16 -> rows {K0,K1}; lane>=16 -> {K2,K3}
    const float* qp = Q + (size_t)col * KK + hi * 2;
    v2f b0, b1;
    b0.x = qp[0]; b0.y = qp[1];
    b1.x = qp[4]; b1.y = qp[5];
    const float al = alpha[col];
    const float be = beta[col];

#pragma unroll
    for (int sub = 0; sub < 4; ++sub)
        wmma_tile16(A, Q, row0 + (sub << 4), lo, hi, b0, b1, al, be, Y, col);
}

__global__ void __launch_bounds__(256) k_gemm_Y_tail(
        const float* __restrict__ A, const float* __restrict__ Q,
        const float* __restrict__ alpha, const float* __restrict__ beta,
        float* __restrict__ Y) {
    const int wave = __builtin_amdgcn_readfirstlane((blockIdx.x << 3) + (threadIdx.x >> 5));
    if (wave >= CTILES) return;                // 125 waves, scalar branch
    const int lane = threadIdx.x & 31;
    const int c0 = wave << 4;
    const int lo = lane & 15;
    const int hi = lane >> 4;
    const int col = c0 + lo;

    const float* qp = Q + (size_t)col * KK + hi * 2;
    v2f b0, b1;
    b0.x = qp[0]; b0.y = qp[1];
    b1.x = qp[4]; b1.y = qp[5];
    const float al = alpha[col];
    const float be = beta[col];

    // rows TAIL_ROW0 .. TAIL_ROW0+15 exactly cover one tile: no guards needed
    wmma_tile16(A, Q, TAIL_ROW0, lo, hi, b0, b1, al, be, Y, col);
}

// ---------------------------------------------------------------------------
// Launch
// ---------------------------------------------------------------------------
extern "C" void kernel_launch(void* const* d_in, const int* in_sizes, int n_in,
                              void* d_out, int out_size, void* d_ws, size_t ws_size,
                              hipStream_t stream) {
    (void)in_sizes; (void)n_in; (void)out_size; (void)ws_size;
    const float* X      = (const float*)d_in[0];
    const int*   cour   = (const int*)  d_in[1];
    const float* CA     = (const float*)d_in[2];
    const float* UA     = (const float*)d_in[3];
    const float* UC     = (const float*)d_in[4];
    const float* guess_ = (const float*)d_in[5];
    const float* slide_ = (const float*)d_in[6];
    float* Y = (float*)d_out;

    // Workspace layout (floats)
    float* ws      = (float*)d_ws;
    float* W       = ws;                                  // C*D*K = 2,048,000
    float* A       = W + (size_t)CC * DD * KK;            // N*K   =   400,000
    float* UAs     = A + (size_t)NN * KK;                 // U*K   =   160,000
    float* rowmax  = UAs + (size_t)UU * KK;               // U     =    20,000
    float* rowinv  = rowmax + UU;                         // U     =    20,000
    float* Qpart   = rowinv + UU;                         // QG*C*K= 1,280,000
    float* Q       = Qpart + (size_t)QG * CC * KK;        // C*K   =    16,000
    float* alpha   = Q + (size_t)CC * KK;                 // C
    float* beta    = alpha + CC;                          // C

    k_softmax_CA   <<<CC, DD, 0, stream>>>(CA, W);
    k_compute_A    <<<NN, DD, 0, stream>>>(X, cour, W, A);
    k_softmax_UA   <<<KK, 256, 0, stream>>>(UA, UAs);
    k_row_stats_UC <<<UU, 256, 0, stream>>>(UC, rowmax, rowinv);
    k_compute_Qpart<<<QG, 256, 0, stream>>>(UC, UAs, rowmax, rowinv, Qpart);
    k_reduce_Q     <<<(CC * KK + 255) / 256, 256, 0, stream>>>(Qpart, Q);
    k_epilogue_consts<<<(CC + 255) / 256, 256, 0, stream>>>(guess_, slide_, alpha, beta);
    k_gemm_Y       <<<(TOTAL_FULL + 7) / 8, 256, 0, stream>>>(A, Q, alpha, beta, Y);
    k_gemm_Y_tail  <<<(CTILES + 7) / 8, 256, 0, stream>>>(A, Q, alpha, beta, Y);
}